// DeformLoss_50208167690302
// MI455X (gfx1250) — compile-verified
//
#include <hip/hip_runtime.h>
#include <hip/hip_bf16.h>
#include <math.h>

// ---------------------------------------------------------------------------
// DeformLoss for MI455X (gfx1250, wave32).
// Chamfer pairwise d2 via V_WMMA_F32_16X16X4_F32:
//   A row i  = (-2x, -2y, -2z, 1)          (16x4, K padded to 4)
//   B col j  = ( tx,  ty,  tz, |t|^2 )     (4x16)
//   C init   = |x_i|^2 broadcast per row   -> D[i][j] = squared distance
// Inner loop: per-lane float2 (global_load_b64, fully coalesced 256B/tile)
// directly forms the B fragment -> no per-iteration selects.
// ---------------------------------------------------------------------------

typedef float v2f __attribute__((ext_vector_type(2)));
typedef float v8f __attribute__((ext_vector_type(8)));

#define BATCH 4
#define KNN_K 6

// ---- workspace accumulator slots (doubles) --------------------------------
// acc[0] = sum of min dist pred->target
// acc[1] = sum of min dist target->pred
// acc[2] = mat sq-sum
// acc[3] = disp sq-sum
// acc[4] = smooth sq-sum

__global__ void dl_init_acc(double* acc) {
    if (threadIdx.x < 8) acc[threadIdx.x] = 0.0;
}

// pack pred: p = template + disp ; out = (p, |p|^2)
__global__ void dl_pack_pred(const float* __restrict__ tpl,
                             const float* __restrict__ disp,
                             float4* __restrict__ out, int total) {
    int i = blockIdx.x * blockDim.x + threadIdx.x;
    if (i >= total) return;
    float x = tpl[3 * i + 0] + disp[3 * i + 0];
    float y = tpl[3 * i + 1] + disp[3 * i + 1];
    float z = tpl[3 * i + 2] + disp[3 * i + 2];
    out[i] = make_float4(x, y, z, x * x + y * y + z * z);
}

// pack target: out = (t, |t|^2)
__global__ void dl_pack_tgt(const float* __restrict__ pos,
                            float4* __restrict__ out, int total) {
    int i = blockIdx.x * blockDim.x + threadIdx.x;
    if (i >= total) return;
    float x = pos[3 * i + 0];
    float y = pos[3 * i + 1];
    float z = pos[3 * i + 2];
    out[i] = make_float4(x, y, z, x * x + y * y + z * z);
}

// One wave owns a 16-row tile of X, sweeps all of Y in 16-col tiles via WMMA,
// keeps per-row min squared distance, then atomically accumulates
// sum(sqrt(max(min_d2, 1e-12))) over its 16 rows.
__global__ void dl_chamfer_min(const float4* __restrict__ X,  // rows, BATCH*nRows
                               const float4* __restrict__ Y,  // cols, BATCH*nCols
                               int nRows, int nCols,
                               double* __restrict__ accum) {
    const int lane = threadIdx.x & 31;
    const int waveInBlock = threadIdx.x >> 5;
    const int wavesPerBlock = blockDim.x >> 5;
    const int wtile = blockIdx.x * wavesPerBlock + waveInBlock;
    const int tilesPerBatch = nRows >> 4;
    const int totalTiles = BATCH * tilesPerBatch;
    if (wtile >= totalTiles) return;          // uniform per wave

    const int b = wtile / tilesPerBatch;
    const int rowTile = wtile - b * tilesPerBatch;
    const int r = (rowTile << 4) + (lane & 15);
    const bool hi = (lane & 16) != 0;         // lane half selects K in {2,3}

    // A fragment (ISA 16x4 f32 layout): v0 = A[r][hi?2:0], v1 = A[r][hi?3:1]
    float4 x4 = X[(size_t)b * nRows + r];
    v2f afrag;
    afrag.x = hi ? (-2.0f * x4.z) : (-2.0f * x4.x);
    afrag.y = hi ? 1.0f           : (-2.0f * x4.y);

    // C init: lanes 0-15 need |x|^2 of rows 0..7, lanes 16-31 rows 8..15
    const int base = hi ? 8 : 0;
    float cinit[8];
#pragma unroll
    for (int v = 0; v < 8; ++v)
        cinit[v] = __shfl(x4.w, base + v, 16);

    float rmin2[8];
#pragma unroll
    for (int v = 0; v < 8; ++v) rmin2[v] = 3.4e38f;

    // Per-lane B-fragment source: lane l<16 reads (y.x,y.y) of column l,
    // lane 16+l reads (y.z,y.w) of column l. One coalesced 256B segment
    // per 16-column tile, one global_load_b64 per lane per tile.
    const float2* __restrict__ yp =
        (const float2*)(Y + (size_t)b * nCols) + 2 * (lane & 15) + (hi ? 1 : 0);

    const int colTiles = nCols >> 4;
    for (int ct = 0; ct < colTiles; ++ct) {
        float2 y2 = yp[(size_t)ct * 32];   // 32 float2 per 16-col tile
        v2f bfrag;
        bfrag.x = y2.x;
        bfrag.y = y2.y;

        v8f c;
#pragma unroll
        for (int v = 0; v < 8; ++v) c[v] = cinit[v];

        c = __builtin_amdgcn_wmma_f32_16x16x4_f32(
                false, afrag, false, bfrag, (short)0, c, false, false);

#pragma unroll
        for (int v = 0; v < 8; ++v) rmin2[v] = fminf(rmin2[v], c[v]);
    }

    // min across the 16 lanes of each half (columns of the D tile)
#pragma unroll
    for (int m = 1; m < 16; m <<= 1) {
#pragma unroll
        for (int v = 0; v < 8; ++v)
            rmin2[v] = fminf(rmin2[v], __shfl_xor(rmin2[v], m, 16));
    }

    // lane 0 carries rows 0-7, lane 16 carries rows 8-15
    if ((lane & 15) == 0) {
        double s = 0.0;
#pragma unroll
        for (int v = 0; v < 8; ++v)
            s += sqrt(fmax((double)rmin2[v], 1e-12));
        atomicAdd(accum, s);
    }
}

// exact top-6 neighbors (excluding self) on template batch 0
__global__ void dl_knn(const float* __restrict__ tpl,
                       int* __restrict__ nn, int n) {
    __shared__ float sx[256], sy[256], sz[256];
    const int i = blockIdx.x * blockDim.x + threadIdx.x;
    float px = 0.f, py = 0.f, pz = 0.f;
    if (i < n) {
        px = tpl[3 * i + 0];
        py = tpl[3 * i + 1];
        pz = tpl[3 * i + 2];
    }
    float bd[KNN_K];
    int   bi[KNN_K];
#pragma unroll
    for (int k = 0; k < KNN_K; ++k) { bd[k] = 3.4e38f; bi[k] = 0; }

    for (int tb = 0; tb < n; tb += 256) {
        int j = tb + threadIdx.x;
        if (j < n) {
            sx[threadIdx.x] = tpl[3 * j + 0];
            sy[threadIdx.x] = tpl[3 * j + 1];
            sz[threadIdx.x] = tpl[3 * j + 2];
        }
        __syncthreads();
        const int lim = min(256, n - tb);
        if (i < n) {
            for (int t = 0; t < lim; ++t) {
                const int jj = tb + t;
                if (jj == i) continue;
                float dx = px - sx[t];
                float dy = py - sy[t];
                float dz = pz - sz[t];
                float d = dx * dx + dy * dy + dz * dz;
                if (d < bd[KNN_K - 1]) {
                    int k = KNN_K - 1;
                    while (k > 0 && bd[k - 1] > d) {
                        bd[k] = bd[k - 1];
                        bi[k] = bi[k - 1];
                        --k;
                    }
                    bd[k] = d;
                    bi[k] = jj;
                }
            }
        }
        __syncthreads();
    }
    if (i < n) {
#pragma unroll
        for (int k = 0; k < KNN_K; ++k) nn[i * KNN_K + k] = bi[k];
    }
}

__global__ void dl_smooth(const float* __restrict__ disp,
                          const int* __restrict__ nn,
                          int n, double* __restrict__ acc) {
    const int idx = blockIdx.x * blockDim.x + threadIdx.x;
    const int total = BATCH * n;
    double s = 0.0;
    if (idx < total) {
        const int b = idx / n;
        const int i = idx - b * n;
        const float* dp = disp + ((size_t)b * n + i) * 3;
        const float dx = dp[0], dy = dp[1], dz = dp[2];
#pragma unroll
        for (int k = 0; k < KNN_K; ++k) {
            const int j = nn[i * KNN_K + k];
            const float* nb = disp + ((size_t)b * n + j) * 3;
            float ex = nb[0] - dx, ey = nb[1] - dy, ez = nb[2] - dz;
            s += (double)(ex * ex + ey * ey + ez * ez);
        }
    }
#pragma unroll
    for (int off = 16; off > 0; off >>= 1)
        s += __shfl_down(s, off, 32);
    if ((threadIdx.x & 31) == 0) atomicAdd(acc, s);
}

// sum (a-b)^2 (b==nullptr -> sum a^2), grid-stride
__global__ void dl_sq_sum(const float* __restrict__ a,
                          const float* __restrict__ bp,
                          int n, double* __restrict__ acc) {
    double s = 0.0;
    for (int i = blockIdx.x * blockDim.x + threadIdx.x; i < n;
         i += gridDim.x * blockDim.x) {
        float d = bp ? (a[i] - bp[i]) : a[i];
        s += (double)d * (double)d;
    }
#pragma unroll
    for (int off = 16; off > 0; off >>= 1)
        s += __shfl_down(s, off, 32);
    if ((threadIdx.x & 31) == 0) atomicAdd(acc, s);
}

__global__ void dl_finalize(const double* __restrict__ acc,
                            float* __restrict__ out, int n, int m) {
    const double cd = 0.5 * (acc[0] / (double)(BATCH * (size_t)n) +
                             acc[1] / (double)(BATCH * (size_t)m));
    const double matL  = acc[2] / (double)((size_t)BATCH * n * 4);
    const double dispL = acc[3] / (double)((size_t)BATCH * n * 3);
    const double smL   = acc[4] / (double)((size_t)BATCH * n * KNN_K * 3);
    out[0] = (float)(1.0 * cd + 0.1 * matL + 0.01 * dispL + 0.005 * smL);
}

extern "C" void kernel_launch(void* const* d_in, const int* in_sizes, int n_in,
                              void* d_out, int out_size, void* d_ws, size_t ws_size,
                              hipStream_t stream) {
    const float* pred_disp  = (const float*)d_in[0];
    const float* pred_mat   = (const float*)d_in[1];
    const float* target_pos = (const float*)d_in[2];
    const float* target_mat = (const float*)d_in[3];
    const float* tpl        = (const float*)d_in[4];

    const int N = in_sizes[0] / (BATCH * 3);
    const int M = in_sizes[2] / (BATCH * 3);

    double* acc   = (double*)d_ws;
    float4* packP = (float4*)((char*)d_ws + 64);
    float4* packT = packP + (size_t)BATCH * N;
    int*    nn    = (int*)(packT + (size_t)BATCH * M);

    dl_init_acc<<<1, 32, 0, stream>>>(acc);

    const int totP = BATCH * N;
    const int totT = BATCH * M;
    dl_pack_pred<<<(totP + 255) / 256, 256, 0, stream>>>(tpl, pred_disp, packP, totP);
    dl_pack_tgt <<<(totT + 255) / 256, 256, 0, stream>>>(target_pos, packT, totT);

    const int wavesPerBlock = 8;  // 256 threads = 8 wave32
    {
        int tiles = BATCH * (N / 16);
        dl_chamfer_min<<<(tiles + wavesPerBlock - 1) / wavesPerBlock,
                         wavesPerBlock * 32, 0, stream>>>(packP, packT, N, M, acc + 0);
    }
    {
        int tiles = BATCH * (M / 16);
        dl_chamfer_min<<<(tiles + wavesPerBlock - 1) / wavesPerBlock,
                         wavesPerBlock * 32, 0, stream>>>(packT, packP, M, N, acc + 1);
    }

    dl_knn<<<(N + 255) / 256, 256, 0, stream>>>(tpl, nn, N);
    dl_smooth<<<(BATCH * N + 255) / 256, 256, 0, stream>>>(pred_disp, nn, N, acc + 4);
    dl_sq_sum<<<256, 256, 0, stream>>>(pred_mat, target_mat, BATCH * N * 4, acc + 2);
    dl_sq_sum<<<256, 256, 0, stream>>>(pred_disp, nullptr, BATCH * N * 3, acc + 3);

    dl_finalize<<<1, 1, 0, stream>>>(acc, (float*)d_out, N, M);
}